// SpatialProjector_51737176047903
// MI455X (gfx1250) — compile-verified
//
#include <hip/hip_runtime.h>

// ---------------------------------------------------------------------------
// SpatialProjector fused kernel for MI455X (gfx1250, wave32, WMMA + async LDS)
//   spatial[b,d,p] = sum_c feat[b,c,d] * mask[b,c,p]        (VALU, LDS tile)
//   y[b,o,p]       = sum_d W[o,d] * spatial[b,d,p] + bias   (V_WMMA_F32_16X16X4_F32)
//   out            = relu(gamma * (y-mean)*rsqrt(var+eps) + beta)  (2-pass BN)
// Mask tile staging uses GLOBAL_LOAD_ASYNC_TO_LDS_B32 (ASYNCcnt) via inline asm.
// Spatial tile stored in WMMA-B-fragment-paired layout so each B fragment is
// one aligned ds_load_b64 (no VGPR reassembly movs before v_wmma).
// ---------------------------------------------------------------------------

typedef float v2f __attribute__((ext_vector_type(2)));
typedef float v8f __attribute__((ext_vector_type(8)));

#define B_   8
#define C_   21
#define D_   64
#define O_   128
#define HW_  65536      // 256*256
#define NT   64         // pixels per workgroup tile (4 WMMA N-subtiles)
#define TILES_PER_B (HW_ / NT)   // 1024
#define BN_EPS 1e-5f

// LDS layout (floats) inside one contiguous block so byte offsets are known:
#define FEAT_OFF 0                       // [C_*D_]   feat[b]
#define MASK_OFF (C_ * D_)               // [C_*NT]   mask tile
#define SP_OFF   (MASK_OFF + C_ * NT)    // [D_*NT]   spatial tile (paired layout)
#define SMEM_FLOATS (SP_OFF + D_ * NT)   // 1344+1344+4096 = 6784 floats (~27 KB)

// Paired spatial layout: float2 element (kg, half, p) holds
//   { sp[4*kg + half][p], sp[4*kg + 2 + half][p] }
// float index of pair start = (kg*(2*NT) + half*NT + p) * 2

// --------------------------- pass 0: zero stats ----------------------------
__global__ void sp_zero_stats(float* __restrict__ ws) {
    int t = threadIdx.x;
    if (t < 2 * O_) ws[t] = 0.0f;
}

// ----------------- pass 1: fused scatter + 1x1 conv (WMMA) -----------------
__global__ __launch_bounds__(256)
void sp_fused_conv(const float* __restrict__ feat,   // [B,C,D]
                   const float* __restrict__ mask,   // [B,C,H,W]
                   const float* __restrict__ Wc,     // [O,D]
                   const float* __restrict__ bias,   // [O]
                   float* __restrict__ y,            // [B,O,H,W] raw (pre-BN)
                   float* __restrict__ sum,          // [O]
                   float* __restrict__ sumsq)        // [O]
{
    __shared__ float smem[SMEM_FLOATS];
    float* s_feat = smem + FEAT_OFF;
    float* s_mask = smem + MASK_OFF;
    float* s_sp   = smem + SP_OFF;

    const int t    = threadIdx.x;
    const int b    = blockIdx.x >> 10;              // / TILES_PER_B
    const int tile = blockIdx.x & (TILES_PER_B - 1);
    const int p0   = tile * NT;

    // ---- stage 0a: async-copy mask tile (21 x 64 f32) global -> LDS ----
    // GVS mode: mem = SADDR(u64) + VADDR(i32) + inst_offset; VDST = LDS byte off
    {
        const float* gbase = mask + ((long long)(b * C_)) * HW_ + p0; // uniform
        for (int i = t; i < C_ * NT; i += 256) {
            const int c = i >> 6;            // / NT
            const int p = i & (NT - 1);
            unsigned goff = (unsigned)((c * HW_ + p) * 4);          // per-lane
            unsigned loff = (unsigned)((MASK_OFF + i) * 4);         // LDS bytes
            asm volatile("global_load_async_to_lds_b32 %0, %1, %2"
                         :: "v"(loff), "v"(goff), "s"(gbase)
                         : "memory");
        }
    }

    // ---- stage 0b: stage feat[b] into LDS with normal loads ----
    for (int i = t; i < C_ * D_; i += 256)
        s_feat[i] = feat[b * C_ * D_ + i];

    // wait for this wave's async copies, then barrier so all waves see LDS
    asm volatile("s_wait_asynccnt 0x0" ::: "memory");
    __syncthreads();

    // ---- stage 1: spatial[d][p] = sum_c feat[c][d] * mask[c][p] ----
    // written in B-fragment-paired layout (see header comment)
    for (int i = t; i < D_ * NT; i += 256) {
        const int d = i >> 6;
        const int p = i & (NT - 1);
        float acc = 0.0f;
#pragma unroll
        for (int c = 0; c < C_; ++c)
            acc = __builtin_fmaf(s_feat[c * D_ + d], s_mask[c * NT + p], acc);

        const int kg   = d >> 2;        // k-group (K step of 4)
        const int r    = d & 3;
        const int half = r & 1;         // which half-wave consumes this row
        const int part = r >> 1;        // 0 -> .x, 1 -> .y of the pair
        s_sp[(kg * (2 * NT) + half * NT + p) * 2 + part] = acc;
    }
    __syncthreads();

    // ---- stage 2: per-wave 16(O) x 64(pix) GEMM via V_WMMA_F32_16X16X4_F32 ----
    const int wave  = t >> 5;          // 0..7  -> O rows [16*wave, 16*wave+16)
    const int lane  = t & 31;
    const int half  = lane >> 4;       // 0: lanes 0-15, 1: lanes 16-31
    const int l16   = lane & 15;
    const int orow0 = wave * 16;

    // per-lane base into paired spatial layout (float2 units)
    const float* spbase = s_sp + (half * NT + l16) * 2;

    v8f cacc[4] = {};                  // 4 N-subtiles of 16x16 f32

#pragma unroll
    for (int k0 = 0; k0 < D_; k0 += 4) {
        // A fragment (16x4, conv_w): lanes 0-15 -> {K0,K1}, lanes 16-31 -> {K2,K3}
        v2f a;
        const int arow = orow0 + l16;
        const int ak   = k0 + 2 * half;
        a.x = Wc[arow * D_ + ak];
        a.y = Wc[arow * D_ + ak + 1];

        const int kg = k0 >> 2;
#pragma unroll
        for (int nt = 0; nt < 4; ++nt) {
            // B fragment: one aligned 8B LDS load straight into a VGPR pair
            v2f bf = *(const v2f*)(spbase + (kg * (2 * NT) + nt * 16) * 2);
            cacc[nt] = __builtin_amdgcn_wmma_f32_16x16x4_f32(
                false, a, false, bf, (short)0, cacc[nt], false, false);
        }
    }

    // ---- stage 3: bias, store raw y, accumulate per-channel BN stats ----
    // C/D layout: VGPR r, lanes 0-15 -> M=r, lanes 16-31 -> M=r+8, N = lane%16
#pragma unroll
    for (int r = 0; r < 8; ++r) {
        const int o  = orow0 + r + 8 * half;
        const float bo = bias[o];
        const long long orow = ((long long)(b * O_ + o)) * HW_ + p0;

        float s = 0.0f, q = 0.0f;
#pragma unroll
        for (int nt = 0; nt < 4; ++nt) {
            float v = cacc[nt][r] + bo;
            y[orow + nt * 16 + l16] = v;
            s += v;
            q += v * v;
        }
        // reduce across the 16 lanes of this half-wave (both halves in parallel)
#pragma unroll
        for (int m = 1; m <= 8; m <<= 1) {
            s += __shfl_xor(s, m, 32);
            q += __shfl_xor(q, m, 32);
        }
        if (l16 == 0) {
            atomicAdd(&sum[o],   s);
            atomicAdd(&sumsq[o], q);
        }
    }
}

// ----------------- pass 2: fold stats into scale/shift ---------------------
__global__ void sp_bn_stats(const float* __restrict__ sum,
                            const float* __restrict__ sumsq,
                            const float* __restrict__ gamma,
                            const float* __restrict__ beta,
                            float* __restrict__ scaleA,
                            float* __restrict__ shiftB)
{
    const int o = threadIdx.x;           // 128 threads
    const float invN = 1.0f / (float)((long long)B_ * HW_);
    const float m   = sum[o] * invN;
    const float var = sumsq[o] * invN - m * m;
    const float inv = rsqrtf(var + BN_EPS);
    const float a   = gamma[o] * inv;
    scaleA[o] = a;
    shiftB[o] = beta[o] - m * a;
}

// ----------------- pass 3: in-place normalize + relu (float4) --------------
__global__ __launch_bounds__(256)
void sp_bn_apply(float* __restrict__ y,
                 const float* __restrict__ scaleA,
                 const float* __restrict__ shiftB)
{
    const long long i = ((long long)blockIdx.x * blockDim.x + threadIdx.x) * 4;
    const int o = (int)((i >> 16) & (O_ - 1));   // (i / HW_) % O_, HW_ = 2^16
    const float a = scaleA[o];
    const float s = shiftB[o];
    float4* p = reinterpret_cast<float4*>(y + i);
    float4 v = *p;
    v.x = fmaxf(__builtin_fmaf(v.x, a, s), 0.0f);
    v.y = fmaxf(__builtin_fmaf(v.y, a, s), 0.0f);
    v.z = fmaxf(__builtin_fmaf(v.z, a, s), 0.0f);
    v.w = fmaxf(__builtin_fmaf(v.w, a, s), 0.0f);
    *p = v;
}

// ---------------------------------------------------------------------------
extern "C" void kernel_launch(void* const* d_in, const int* in_sizes, int n_in,
                              void* d_out, int out_size, void* d_ws, size_t ws_size,
                              hipStream_t stream) {
    const float* feat  = (const float*)d_in[0];  // node_features [B,C,D]
    const float* mask  = (const float*)d_in[1];  // node_masks    [B,C,H,W]
    const float* Wc    = (const float*)d_in[2];  // conv_w        [O,D]
    const float* bias  = (const float*)d_in[3];  // conv_b        [O]
    const float* gamma = (const float*)d_in[4];  // gamma         [O]
    const float* beta  = (const float*)d_in[5];  // beta          [O]
    float* y = (float*)d_out;                    // [B,O,H,W]

    float* ws     = (float*)d_ws;
    float* sum    = ws;            // [128]
    float* sumsq  = ws + O_;       // [128]
    float* scaleA = ws + 2 * O_;   // [128]
    float* shiftB = ws + 3 * O_;   // [128]

    sp_zero_stats<<<1, 256, 0, stream>>>(ws);

    const int nblocks1 = B_ * TILES_PER_B;       // 8192
    sp_fused_conv<<<nblocks1, 256, 0, stream>>>(feat, mask, Wc, bias, y, sum, sumsq);

    sp_bn_stats<<<1, O_, 0, stream>>>(sum, sumsq, gamma, beta, scaleA, shiftB);

    const long long nelem = (long long)B_ * O_ * HW_;   // 67,108,864
    const int nblocks3 = (int)(nelem / (4 * 256));      // 65536
    sp_bn_apply<<<nblocks3, 256, 0, stream>>>(y, scaleA, shiftB);
}